// RGCLayer_85650237817477
// MI455X (gfx1250) — compile-verified
//
#include <hip/hip_runtime.h>

// RGCN basis-decomposition layer for MI455X (gfx1250, wave32).
//   N=20000 nodes, R=4 relations, B=4 bases, H=128, E=160000 edges/relation.
//
// Pipeline:
//   1) zero d_out (atomic accumulation target)
//   2) Vf(N*R x H) = basis combine via V_WMMA_F32_16X16X4_F32  (K = B = 4)
//   3) edge scatter: gather Vf row (r*N+dst), scale, GLOBAL_ATOMIC_ADD_F32 to out[src]
//   4) ReLU in place
//
// Roofline: basis combine = 82 MFLOP over ~82 MB (trivially BW-bound, ~4us at
// 23.3 TB/s). Scatter dominates: 640K x 512B gathers from a 41 MB Vf (L2-resident,
// 192 MB L2) + 640K x 128 f32 atomics into a 10 MB output (L2-resident).

typedef __attribute__((ext_vector_type(2))) float v2f;
typedef __attribute__((ext_vector_type(8))) float v8f;

#define R_REL   4
#define B_BASES 4
#define H_DIM   128

// Hardware f32 atomic add, no return (STOREcnt-tracked; implicit wait at s_endpgm).
static __device__ __forceinline__ void atomic_add_f32_hw(float* p, float v) {
  asm volatile("global_atomic_add_f32 %0, %1, off" : : "v"(p), "v"(v) : "memory");
}

// ---------------- Kernel 1: zero the output ----------------
__global__ void rgcn_zero_f4(float4* __restrict__ out, int n4) {
  int i = blockIdx.x * blockDim.x + threadIdx.x;
  if (i < n4) out[i] = make_float4(0.f, 0.f, 0.f, 0.f);
}

// ---------------- Kernel 2: basis combine via WMMA ----------------
// Y(16 x 16-tile) = A(16x4, W_comp zero-padded) * B(4x16, slice of W viewed (B, N*H))
// A layout (32-bit A 16x4): lanes 0-15: M=0..15, v0=K0, v1=K1; lanes 16-31: v0=K2, v1=K3.
// B layout (4x16): lane n (0-15): v0=B[K0,n], v1=B[K1,n]; lane n+16: v0=B[K2,n], v1=B[K3,n].
// D layout: VGPR v, lanes 0-15 -> row M=v. Only rows 0..3 (the 4 relations) are useful.
__global__ void rgcn_basis_wmma(const float* __restrict__ W,   // (B, N*H) contiguous view
                                const float* __restrict__ Wc,  // (R, B) row-major
                                float* __restrict__ Vf,        // (N*R, H) node-major
                                int NH) {
  const int lane = threadIdx.x & 31;
  const int wave = blockIdx.x * (blockDim.x >> 5) + (threadIdx.x >> 5);
  const int nw   = gridDim.x * (blockDim.x >> 5);
  const int M    = lane & 15;
  const int kp   = lane >> 4;  // 0: K=0,1   1: K=2,3

  v2f a;
  a.x = (M < R_REL) ? Wc[M * B_BASES + 2 * kp + 0] : 0.0f;
  a.y = (M < R_REL) ? Wc[M * B_BASES + 2 * kp + 1] : 0.0f;

  const int ntiles = NH >> 4;
  for (int t = wave; t < ntiles; t += nw) {   // wave-uniform loop (EXEC all-1 at wmma)
    const int j0  = t << 4;
    const int col = j0 + M;
    v2f b;
    b.x = W[(size_t)(2 * kp + 0) * NH + col];
    b.y = W[(size_t)(2 * kp + 1) * NH + col];
    v8f c = {};
    // (neg_a, A, neg_b, B, c_mod, C, reuse_a, reuse_b)
    c = __builtin_amdgcn_wmma_f32_16x16x4_f32(false, a, false, b, (short)0, c,
                                              false, false);
    if (lane < 16) {
      // j0 -> node n = j0/H, h0 = j0%H (tile never crosses a node: 16 | H)
      const int n  = j0 >> 7;          // H_DIM == 128
      const int h0 = j0 & (H_DIM - 1);
      float* dst = Vf + (size_t)n * (R_REL * H_DIM) + h0 + lane;
      dst[0 * H_DIM] = c[0];   // r = 0
      dst[1 * H_DIM] = c[1];   // r = 1
      dst[2 * H_DIM] = c[2];   // r = 2
      dst[3 * H_DIM] = c[3];   // r = 3
    }
  }
}

// ---------------- Kernel 3: edge scatter with f32 global atomics ----------------
// One wave per edge; lane handles 4 of the 128 columns (float4 gather, 4 atomic adds).
__global__ void rgcn_scatter(const int*   __restrict__ esrc,
                             const int*   __restrict__ edst,
                             const float* __restrict__ evals,
                             const float* __restrict__ Vf,
                             float*       __restrict__ out,
                             int E, int N) {
  const int lane = threadIdx.x & 31;
  const int e    = blockIdx.x * (blockDim.x >> 5) + (threadIdx.x >> 5);
  const int r    = blockIdx.y;
  if (e >= E) return;

  const int idx = r * E + e;          // (R, E) flat
  const int src = esrc[idx];
  const int dst = edst[idx];
  const float v = evals[idx];

  const int c = r * N + dst;          // faithful quirk: relation-major index into
                                      // node-major Vf, exactly like the reference
  const float4 x = *(const float4*)(Vf + (size_t)c * H_DIM + lane * 4);
  float* o = out + (size_t)src * H_DIM + lane * 4;
  atomic_add_f32_hw(o + 0, v * x.x);  // -> global_atomic_add_f32 (guaranteed, no CAS)
  atomic_add_f32_hw(o + 1, v * x.y);
  atomic_add_f32_hw(o + 2, v * x.z);
  atomic_add_f32_hw(o + 3, v * x.w);
}

// ---------------- Kernel 4: ReLU in place ----------------
__global__ void rgcn_relu_f4(float4* __restrict__ out, int n4) {
  int i = blockIdx.x * blockDim.x + threadIdx.x;
  if (i < n4) {
    float4 x = out[i];
    x.x = fmaxf(x.x, 0.f);
    x.y = fmaxf(x.y, 0.f);
    x.z = fmaxf(x.z, 0.f);
    x.w = fmaxf(x.w, 0.f);
    out[i] = x;
  }
}

extern "C" void kernel_launch(void* const* d_in, const int* in_sizes, int n_in,
                              void* d_out, int out_size, void* d_ws, size_t ws_size,
                              hipStream_t stream) {
  const int*   esrc  = (const int*)  d_in[0];   // (R, E) int32
  const int*   edst  = (const int*)  d_in[1];   // (R, E) int32
  const float* evals = (const float*)d_in[2];   // (R, E) f32
  const float* W     = (const float*)d_in[3];   // (N*B, H) f32  == (B, N*H) view
  const float* Wc    = (const float*)d_in[4];   // (R, B) f32
  float* out = (float*)d_out;                   // (N, H) f32
  float* Vf  = (float*)d_ws;                    // (N*R, H) f32 scratch: 40.96 MB

  const int E  = in_sizes[0] / R_REL;                       // 160000
  const int N  = in_sizes[3] / (B_BASES * H_DIM);           // 20000
  const int NH = N * H_DIM;                                 // 2,560,000

  // 1) zero output (atomic target; harness poisons buffers)
  const int n4 = NH / 4;
  rgcn_zero_f4<<<(n4 + 255) / 256, 256, 0, stream>>>((float4*)out, n4);

  // 2) basis combine: 160K 16-col tiles, 8192 waves grid-striding
  rgcn_basis_wmma<<<1024, 256, 0, stream>>>(W, Wc, Vf, NH);

  // 3) scatter: one wave per edge, grid.y = relation
  dim3 sgrid((E + 7) / 8, R_REL);
  rgcn_scatter<<<sgrid, 256, 0, stream>>>(esrc, edst, evals, Vf, out, E, N);

  // 4) ReLU
  rgcn_relu_f4<<<(n4 + 255) / 256, 256, 0, stream>>>((float4*)out, n4);
}